// RandomPooling2D_10024453669361
// MI455X (gfx1250) — compile-verified
//
#include <hip/hip_runtime.h>

// out[b,c,i,j] = x[b,c, 2*i + pick_i, 2*j + pick_j]
// B=16, C=64, H=W=512, OH=OW=256. Pure streaming gather: ~512MB read + 256MB
// write -> HBM-bound (~33us at 23.3 TB/s). Strategy: 128-bit coalesced loads
// and stores with non-temporal (TH=NT) cache policy since both streams exceed
// the 192MB L2.

typedef __attribute__((ext_vector_type(4))) float v4f;

__global__ __launch_bounds__(256) void RandomPooling2D_slice_kernel(
    const float* __restrict__ x,
    const int* __restrict__ pick_i_p,
    const int* __restrict__ pick_j_p,
    float* __restrict__ out)
{
    const int pick_i = pick_i_p[0];   // runtime-uniform scalar
    const int pick_j = pick_j_p[0];   // runtime-uniform scalar

    // One thread per output float4. Total out vec4s = 16*64*256*256/4 = 2^24.
    const unsigned idx  = blockIdx.x * 256u + threadIdx.x;
    const unsigned j4   = idx & 63u;        // vec4 index along OW (OW/4 = 64)
    const unsigned rest = idx >> 6;
    const unsigned i    = rest & 255u;      // output row, OH = 256
    const unsigned bc   = rest >> 8;        // fused (b,c), 0..1023

    // Input row: (bc * H + 2*i + pick_i) * W floats from x.
    const size_t in_row = ((size_t)bc * 512u + (2u * i + (unsigned)pick_i)) * 512u;

    // 8 consecutive input floats starting at 8*j4 cover output elements
    // 4*j4 .. 4*j4+3 for either pick_j in {0,1}. Both loads are 16B-aligned.
    const v4f* pin = (const v4f*)(x + in_row + 8u * (size_t)j4);
    v4f a = __builtin_nontemporal_load(pin);        // input floats 8j4 .. 8j4+3
    v4f b = __builtin_nontemporal_load(pin + 1);    // input floats 8j4+4 .. 8j4+7

    v4f o;
    if (pick_j == 0) {
        o = (v4f){a.x, a.z, b.x, b.z};
    } else {
        o = (v4f){a.y, a.w, b.y, b.w};
    }

    __builtin_nontemporal_store(o, (v4f*)out + idx);
}

extern "C" void kernel_launch(void* const* d_in, const int* in_sizes, int n_in,
                              void* d_out, int out_size, void* d_ws, size_t ws_size,
                              hipStream_t stream) {
    const float* x  = (const float*)d_in[0];
    const int*   pi = (const int*)d_in[1];
    const int*   pj = (const int*)d_in[2];
    float*       out = (float*)d_out;

    // out_size = 16*64*256*256 = 2^26 floats -> 2^24 float4 -> 65536 blocks x 256 threads
    const unsigned n_vec4 = 16u * 64u * 256u * 256u / 4u;
    dim3 grid(n_vec4 / 256u);
    dim3 block(256);
    hipLaunchKernelGGL(RandomPooling2D_slice_kernel, grid, block, 0, stream,
                       x, pi, pj, out);
}